// GDN_64089501990915
// MI455X (gfx1250) — compile-verified
//
#include <hip/hip_runtime.h>
#include <math.h>

#define BATCH 256
#define NODE  256
#define IN_DIM 64
#define DIM   64
#define TOPK  20
#define BN    (BATCH*NODE)
#define EPS   1e-5f
#define NEG_SLOPE 0.2f

typedef float v2f __attribute__((ext_vector_type(2)));
typedef float v8f __attribute__((ext_vector_type(8)));

// ---------------- init: zero split-K accumulator + BN statistics ----------------
__global__ void k_init(float* __restrict__ enc, float* __restrict__ stats) {
  int i = blockIdx.x * 256 + threadIdx.x;
  if (i < BATCH * DIM) enc[i] = 0.f;
  if (i < 256) stats[i] = 0.f;
}

// ---------------- top-k cosine neighbors (one block per row) ----------------
__global__ void k_topk(const float* __restrict__ emb, int* __restrict__ topk) {
  __shared__ float wi[DIM];
  __shared__ float cosv[NODE];
  __shared__ float redv[NODE];
  __shared__ int   redi[NODE];
  int i = blockIdx.x, j = threadIdx.x;
  if (j < DIM) wi[j] = emb[i * DIM + j];
  __syncthreads();
  float dot = 0.f, nj = 0.f, ni = 0.f;
  for (int d = 0; d < DIM; ++d) {
    float a = wi[d];
    float b = emb[j * DIM + d];
    dot += a * b; nj += b * b; ni += a * a;
  }
  cosv[j] = dot / (sqrtf(ni) * sqrtf(nj));
  __syncthreads();
  for (int k = 0; k < TOPK; ++k) {
    redv[j] = cosv[j]; redi[j] = j;
    __syncthreads();
    for (int s = 128; s > 0; s >>= 1) {
      if (j < s) {
        float va = redv[j], vb = redv[j + s];
        int ia = redi[j], ib = redi[j + s];
        if (vb > va || (vb == va && ib < ia)) { redv[j] = vb; redi[j] = ib; }
      }
      __syncthreads();
    }
    if (j == 0) { topk[i * TOPK + k] = redi[0]; cosv[redi[0]] = -1e30f; }
    __syncthreads();
  }
}

// ---------------- xl = x @ lin_w  (fp32 WMMA 16x16x4), 1 wave = 16x64 tile ----------------
__global__ void k_lin_wmma(const float* __restrict__ x, const float* __restrict__ w,
                           float* __restrict__ xl) {
  int wave = threadIdx.x >> 5;
  int lane = threadIdx.x & 31;
  int half = lane >> 4, l16 = lane & 15;
  int m_base = (blockIdx.x * 4 + wave) * 16;
  const float* arow = x + (size_t)(m_base + l16) * IN_DIM;
  v8f c0 = {}, c1 = {}, c2 = {}, c3 = {};
  for (int k0 = 0; k0 < IN_DIM; k0 += 4) {
    int kk = k0 + 2 * half;
    v2f a; a.x = arow[kk]; a.y = arow[kk + 1];
    const float* b0p = w + (size_t)kk * DIM + l16;
    const float* b1p = b0p + DIM;
    v2f b;
    b.x = b0p[0];  b.y = b1p[0];
    c0 = __builtin_amdgcn_wmma_f32_16x16x4_f32(false, a, false, b, 0, c0, false, false);
    b.x = b0p[16]; b.y = b1p[16];
    c1 = __builtin_amdgcn_wmma_f32_16x16x4_f32(false, a, false, b, 0, c1, false, false);
    b.x = b0p[32]; b.y = b1p[32];
    c2 = __builtin_amdgcn_wmma_f32_16x16x4_f32(false, a, false, b, 0, c2, false, false);
    b.x = b0p[48]; b.y = b1p[48];
    c3 = __builtin_amdgcn_wmma_f32_16x16x4_f32(false, a, false, b, 0, c3, false, false);
  }
  for (int r = 0; r < 8; ++r) {
    int row = m_base + r + 8 * half;
    float* o = xl + (size_t)row * DIM + l16;
    o[0] = c0[r]; o[16] = c1[r]; o[32] = c2[r]; o[48] = c3[r];
  }
}

// ---------------- per-node attention scalars ----------------
__global__ void k_scalars(const float* __restrict__ xl, const float* __restrict__ emb,
                          const float* __restrict__ ai, const float* __restrict__ aj,
                          const float* __restrict__ aei, const float* __restrict__ aej,
                          float* __restrict__ tdst, float* __restrict__ tsrc) {
  int n = blockIdx.x * 256 + threadIdx.x;
  const float* xr = xl + (size_t)n * DIM;
  const float* er = emb + (size_t)(n & (NODE - 1)) * DIM;
  float si = 0.f, sj = 0.f, ei = 0.f, ej = 0.f;
  for (int d = 0; d < DIM; ++d) {
    float xv = xr[d], ev = er[d];
    si += xv * ai[d]; sj += xv * aj[d];
    ei += ev * aei[d]; ej += ev * aej[d];
  }
  tdst[n] = si + ei;
  tsrc[n] = sj + ej;
}

// ---------------- attention softmax + weighted aggregation (4 targets / block) ----------------
__global__ void k_attn_agg(const float* __restrict__ xl, const int* __restrict__ topk,
                           const float* __restrict__ tdst, const float* __restrict__ tsrc,
                           const float* __restrict__ gbias, float* __restrict__ agg) {
  __shared__ float sh_logit[4][24];
  __shared__ int   sh_src[4][24];
  __shared__ float sh_alpha[4][24];
  int t = threadIdx.x;
  int slot = t >> 6, d = t & 63;
  int tgt = blockIdx.x * 4 + slot;
  int i = tgt & (NODE - 1);
  int b = tgt >> 8;
  if (d < TOPK + 1) {
    int srcn; float logit;
    if (d < TOPK) {
      int s = topk[i * TOPK + d];
      srcn = b * NODE + s;
      float raw = tdst[tgt] + tsrc[srcn];
      float lr = raw >= 0.f ? raw : NEG_SLOPE * raw;
      logit = (s != i) ? lr : -1e9f;   // self edges from top-k are masked
    } else {
      srcn = tgt;                      // appended self loop (always valid)
      float raw = tdst[tgt] + tsrc[tgt];
      logit = raw >= 0.f ? raw : NEG_SLOPE * raw;
    }
    sh_logit[slot][d] = logit;
    sh_src[slot][d] = srcn;
  }
  __syncthreads();
  if (d == 0) {
    float m = -1e30f;
    for (int e = 0; e < TOPK + 1; ++e) m = fmaxf(m, sh_logit[slot][e]);
    float ex[TOPK + 1];
    float den = 0.f;
    for (int e = 0; e < TOPK + 1; ++e) { ex[e] = expf(sh_logit[slot][e] - m); den += ex[e]; }
    float inv = 1.f / den;
    for (int e = 0; e < TOPK + 1; ++e) sh_alpha[slot][e] = ex[e] * inv;
  }
  __syncthreads();
  float acc = gbias[d];
  for (int e = 0; e < TOPK + 1; ++e)
    acc += sh_alpha[slot][e] * xl[(size_t)sh_src[slot][e] * DIM + d];
  agg[(size_t)tgt * DIM + d] = acc;
}

// ---------------- BN1 statistics: per-channel sum / sumsq over BN rows ----------------
__global__ void k_reduce_bn1(const float* __restrict__ agg, float* __restrict__ stats) {
  __shared__ float ssum[256], ssq[256];
  int t = threadIdx.x;
  int d = t & 63, rg = t >> 6;
  int rowbase = blockIdx.x * 256 + rg;
  float s = 0.f, q = 0.f;
  for (int k = 0; k < 64; ++k) {
    float v = agg[(size_t)(rowbase + 4 * k) * DIM + d];
    s += v; q += v * v;
  }
  ssum[t] = s; ssq[t] = q;
  __syncthreads();
  if (t < 128) { ssum[t] += ssum[t + 128]; ssq[t] += ssq[t + 128]; }
  __syncthreads();
  if (t < 64) {
    ssum[t] += ssum[t + 64]; ssq[t] += ssq[t + 64];
    atomicAdd(&stats[t], ssum[t]);
    atomicAdd(&stats[64 + t], ssq[t]);
  }
}

// ---------------- apply BN1+ReLU in place, fused BN2 statistics on y = gnn*emb ----------------
__global__ void k_bn1_bn2red(float* __restrict__ gnn, float* __restrict__ stats,
                             const float* __restrict__ g1, const float* __restrict__ b1,
                             const float* __restrict__ emb) {
  __shared__ float ssum[256], ssq[256];
  int t = threadIdx.x;
  size_t e = (size_t)blockIdx.x * 256 + t;
  int d = t & 63;
  int node = (int)(e >> 6);
  float mean = stats[d] * (1.f / BN);
  float var  = stats[64 + d] * (1.f / BN) - mean * mean;
  float v = (gnn[e] - mean) * rsqrtf(var + EPS) * g1[d] + b1[d];
  v = fmaxf(v, 0.f);
  gnn[e] = v;
  float y = v * emb[(size_t)(node & (NODE - 1)) * DIM + d];
  ssum[t] = y; ssq[t] = y * y;
  __syncthreads();
  if (t < 128) { ssum[t] += ssum[t + 128]; ssq[t] += ssq[t + 128]; }
  __syncthreads();
  if (t < 64) {
    ssum[t] += ssum[t + 64]; ssq[t] += ssq[t + 64];
    atomicAdd(&stats[128 + t], ssum[t]);
    atomicAdd(&stats[192 + t], ssq[t]);
  }
}

// ---------------- encoded = gnn.view(256,16384) @ enc_w : fp32 WMMA split-K ----------------
__global__ void k_enc_wmma(const float* __restrict__ G, const float* __restrict__ W,
                           float* __restrict__ enc) {
  int lane = threadIdx.x & 31;
  int half = lane >> 4, l16 = lane & 15;
  int m_base = blockIdx.x * 16;
  int k_base = blockIdx.y * 128;
  const float* arow = G + (size_t)(m_base + l16) * (NODE * DIM);
  v8f c0 = {}, c1 = {}, c2 = {}, c3 = {};
  for (int k0 = k_base; k0 < k_base + 128; k0 += 4) {
    int kk = k0 + 2 * half;
    v2f a; a.x = arow[kk]; a.y = arow[kk + 1];
    const float* b0p = W + (size_t)kk * DIM + l16;
    const float* b1p = b0p + DIM;
    v2f b;
    b.x = b0p[0];  b.y = b1p[0];
    c0 = __builtin_amdgcn_wmma_f32_16x16x4_f32(false, a, false, b, 0, c0, false, false);
    b.x = b0p[16]; b.y = b1p[16];
    c1 = __builtin_amdgcn_wmma_f32_16x16x4_f32(false, a, false, b, 0, c1, false, false);
    b.x = b0p[32]; b.y = b1p[32];
    c2 = __builtin_amdgcn_wmma_f32_16x16x4_f32(false, a, false, b, 0, c2, false, false);
    b.x = b0p[48]; b.y = b1p[48];
    c3 = __builtin_amdgcn_wmma_f32_16x16x4_f32(false, a, false, b, 0, c3, false, false);
  }
  for (int r = 0; r < 8; ++r) {
    int row = m_base + r + 8 * half;
    float* o = enc + (size_t)row * DIM + l16;
    atomicAdd(&o[0],  c0[r]);
    atomicAdd(&o[16], c1[r]);
    atomicAdd(&o[32], c2[r]);
    atomicAdd(&o[48], c3[r]);
  }
}

// ---------------- arrangement = (encoded + enc_b) @ arr_w + arr_b ----------------
__global__ void k_arr(const float* __restrict__ enc, const float* __restrict__ encb,
                      const float* __restrict__ aw, const float* __restrict__ ab,
                      float* __restrict__ out_arr) {
  int b = threadIdx.x;
  float e[DIM];
  for (int d = 0; d < DIM; ++d) e[d] = enc[b * DIM + d] + encb[d];
  for (int o = 0; o < 7; ++o) {
    float s = ab[o];
    for (int d = 0; d < DIM; ++d) s += e[d] * aw[d * 7 + o];
    out_arr[b * 7 + o] = s;
  }
}

// ---------------- BN2 apply + ReLU + [64x1] projection -> score ----------------
__global__ void k_score(const float* __restrict__ gnn, const float* __restrict__ emb,
                        const float* __restrict__ stats, const float* __restrict__ g2,
                        const float* __restrict__ b2, const float* __restrict__ ow,
                        const float* __restrict__ ob, float* __restrict__ score) {
  int t = blockIdx.x * 256 + threadIdx.x;   // flat (b, n) index
  int n = t & (NODE - 1);
  float acc = 0.f;
  for (int d = 0; d < DIM; ++d) {
    float mean = stats[128 + d] * (1.f / BN);
    float var  = stats[192 + d] * (1.f / BN) - mean * mean;
    float y = gnn[(size_t)t * DIM + d] * emb[n * DIM + d];
    float v = (y - mean) * rsqrtf(var + EPS) * g2[d] + b2[d];
    acc += fmaxf(v, 0.f) * ow[d];
  }
  score[t] = acc + ob[0];
}

extern "C" void kernel_launch(void* const* d_in, const int* in_sizes, int n_in,
                              void* d_out, int out_size, void* d_ws, size_t ws_size,
                              hipStream_t stream) {
  const float* data   = (const float*)d_in[0];
  const float* emb    = (const float*)d_in[1];
  const float* lin_w  = (const float*)d_in[2];
  const float* att_i  = (const float*)d_in[3];
  const float* att_j  = (const float*)d_in[4];
  const float* att_em_i = (const float*)d_in[5];
  const float* att_em_j = (const float*)d_in[6];
  const float* gnn_bias = (const float*)d_in[7];
  const float* bn1g   = (const float*)d_in[8];
  const float* bn1b   = (const float*)d_in[9];
  const float* bn2g   = (const float*)d_in[10];
  const float* bn2b   = (const float*)d_in[11];
  const float* enc_w  = (const float*)d_in[12];
  const float* enc_b  = (const float*)d_in[13];
  const float* arr_w  = (const float*)d_in[14];
  const float* arr_b  = (const float*)d_in[15];
  const float* out_w  = (const float*)d_in[16];
  const float* out_b  = (const float*)d_in[17];
  float* out = (float*)d_out;

  char* ws = (char*)d_ws;
  size_t off = 0;
  auto alloc = [&](size_t bytes) -> char* {
    char* p = ws + off;
    off += (bytes + 255) & ~(size_t)255;
    return p;
  };
  int*   topk = (int*)  alloc((size_t)NODE * TOPK * sizeof(int));
  float* xl   = (float*)alloc((size_t)BN * DIM * sizeof(float));
  float* agg  = (float*)alloc((size_t)BN * DIM * sizeof(float));  // becomes gnn_out in place
  float* tdst = (float*)alloc((size_t)BN * sizeof(float));
  float* tsrc = (float*)alloc((size_t)BN * sizeof(float));
  float* enc  = (float*)alloc((size_t)BATCH * DIM * sizeof(float));
  float* stats= (float*)alloc(256 * sizeof(float));

  k_init<<<65, 256, 0, stream>>>(enc, stats);
  k_topk<<<NODE, 256, 0, stream>>>(emb, topk);
  k_lin_wmma<<<BN / 64, 128, 0, stream>>>(data, lin_w, xl);
  k_scalars<<<256, 256, 0, stream>>>(xl, emb, att_i, att_j, att_em_i, att_em_j, tdst, tsrc);
  k_attn_agg<<<BN / 4, 256, 0, stream>>>(xl, topk, tdst, tsrc, gnn_bias, agg);
  k_reduce_bn1<<<256, 256, 0, stream>>>(agg, stats);
  k_bn1_bn2red<<<(BN * DIM) / 256, 256, 0, stream>>>(agg, stats, bn1g, bn1b, emb);
  k_enc_wmma<<<dim3(16, 128), 32, 0, stream>>>(agg, enc_w, enc);
  k_arr<<<1, 256, 0, stream>>>(enc, enc_b, arr_w, arr_b, out + BN);
  k_score<<<256, 256, 0, stream>>>(agg, emb, stats, bn2g, bn2b, out_w, out_b, out);
}